// Interpolater_89258010346078
// MI455X (gfx1250) — compile-verified
//
#include <hip/hip_runtime.h>
#include <math.h>

typedef __attribute__((ext_vector_type(2))) float v2f;
typedef __attribute__((ext_vector_type(8))) float v8f;

#define EPSF 1e-7f
#define BASE_SCALE_F 0.02f
#define SENTINEL 1e30f

// Branchless sorted top-4 (smallest) insertion: s0<=s1<=s2<=s3.
__device__ __forceinline__ void top4_insert(float c, int ci,
    float& s0, float& s1, float& s2, float& s3,
    int& i0, int& i1, int& i2, int& i3)
{
    bool b; float tf; int ti;
    b = c < s3;  s3 = b ? c  : s3; i3 = b ? ci : i3;
    b = s3 < s2; tf = s2; ti = i2;
    s2 = b ? s3 : s2; i2 = b ? i3 : i2;
    s3 = b ? tf : s3; i3 = b ? ti : i3;
    b = s2 < s1; tf = s1; ti = i1;
    s1 = b ? s2 : s1; i1 = b ? i2 : i1;
    s2 = b ? tf : s2; i2 = b ? ti : i2;
    b = s1 < s0; tf = s0; ti = i0;
    s0 = b ? s1 : s0; i0 = b ? i1 : i0;
    s1 = b ? tf : s1; i1 = b ? ti : i1;
}

// Tail insert: caller guarantees c < s3 (used under a divergent guard so the
// wave can skip the whole bubble with s_cbranch_execz in the common case).
__device__ __forceinline__ void top4_insert_tail(float c, int ci,
    float& s0, float& s1, float& s2, float& s3,
    int& i0, int& i1, int& i2, int& i3)
{
    bool b; float tf; int ti;
    s3 = c; i3 = ci;
    b = s3 < s2; tf = s2; ti = i2;
    s2 = b ? s3 : s2; i2 = b ? i3 : i2;
    s3 = b ? tf : s3; i3 = b ? ti : i3;
    b = s2 < s1; tf = s1; ti = i1;
    s1 = b ? s2 : s1; i1 = b ? i2 : i1;
    s2 = b ? tf : s2; i2 = b ? ti : i2;
    b = s1 < s0; tf = s0; ti = i0;
    s0 = b ? s1 : s0; i0 = b ? i1 : i0;
    s1 = b ? tf : s1; i1 = b ? ti : i1;
}

// ---------------------------------------------------------------------------
// Kernel 1: loss_disp partial sums (per-vertex ||displacement||)
// ---------------------------------------------------------------------------
__global__ void __launch_bounds__(256) prep_disp_kernel(
    const float* __restrict__ disp, int V, float* __restrict__ part)
{
    __shared__ float red[256];
    int v = blockIdx.x * 256 + threadIdx.x;
    float l = 0.0f;
    if (v < V) {
        float a = disp[3 * v + 0];
        float b = disp[3 * v + 1];
        float c = disp[3 * v + 2];
        l = sqrtf(a * a + b * b + c * c);
    }
    red[threadIdx.x] = l;
    __syncthreads();
    for (int s = 128; s > 0; s >>= 1) {
        if (threadIdx.x < s) red[threadIdx.x] += red[threadIdx.x + s];
        __syncthreads();
    }
    if (threadIdx.x == 0) part[blockIdx.x] = red[0];
}

// ---------------------------------------------------------------------------
// Kernel 2: fused KNN (WMMA f32 16x16x4 score tiles) + interpolation + losses
//   block = 256 threads = 8 waves; each wave owns 16 gaussians (WMMA columns)
//   dynamic LDS: (Vpad + 16) * float4 {x, y, z, |v|^2}; +16 = prefetch pad
// ---------------------------------------------------------------------------
__global__ void __launch_bounds__(256) nn_interp_kernel(
    const float* __restrict__ g_xyz, const float* __restrict__ v_xyz,
    const float* __restrict__ disp,  const float* __restrict__ vsc,
    const float* __restrict__ vrot,  const float* __restrict__ lbs,
    const float* __restrict__ vnrm,
    int N, int V, int Vpad,
    float* __restrict__ out, float* __restrict__ part)
{
    extern __shared__ float s_vert[];   // (Vpad + 16) * 4 floats
    __shared__ float s_red[24];         // 8 waves * 3 loss partials

    const int tid  = threadIdx.x;
    const int lane = tid & 31;
    const int wid  = tid >> 5;
    const int half = lane >> 4;         // 0: lanes 0-15, 1: lanes 16-31
    const int m    = lane & 15;

    // Stage vertices into LDS as [x, y, z, |v|^2]; pad rows get huge score.
    for (int i = tid; i < Vpad; i += 256) {
        float x = 0.0f, y = 0.0f, z = 0.0f, n2 = SENTINEL;
        if (i < V) {
            x = v_xyz[3 * i + 0];
            y = v_xyz[3 * i + 1];
            z = v_xyz[3 * i + 2];
            n2 = x * x + y * y + z * z;
        }
        s_vert[4 * i + 0] = x;
        s_vert[4 * i + 1] = y;
        s_vert[4 * i + 2] = z;
        s_vert[4 * i + 3] = n2;
    }
    __syncthreads();

    // B matrix (4x16 f32): column = gaussian (N = lane%16).
    // lanes 0-15 hold K0,K1 = {-2gx, -2gy}; lanes 16-31 hold K2,K3 = {-2gz, 1}.
    const int gbase = blockIdx.x * 128 + wid * 16;
    const int gclmp = (gbase + m < N) ? (gbase + m) : (N - 1);
    const float gx = g_xyz[3 * gclmp + 0];
    const float gy = g_xyz[3 * gclmp + 1];
    const float gz = g_xyz[3 * gclmp + 2];
    v2f Bv;
    Bv[0] = half ? (-2.0f * gz) : (-2.0f * gx);
    Bv[1] = half ? 1.0f         : (-2.0f * gy);

    // Per-lane top-4 of scores = -2 g.v + |v|^2 (|g|^2 is order-invariant).
    float s0 = 3e30f, s1 = 3e30f, s2 = 3e30f, s3 = 3e30f;
    int   i0 = 0, i1 = 0, i2 = 0, i3 = 0;

    const int ntiles = Vpad >> 4;

    // A matrix (16x4 f32): row = vertex (M = lane%16).
    // lanes 0-15 read {x,y}; lanes 16-31 read {z,|v|^2}: one ds_load_b64.
    // Software-pipelined: prefetch tile t+1 before the WMMA of tile t.
    const float* vp = &s_vert[(m << 2) + (half << 1)];
    v2f Av = *(const v2f*)vp;
    for (int t = 0; t < ntiles; ++t) {
        v2f Acur = Av;
        vp += 64;                       // next 16-vertex tile (256 B)
        Av = *(const v2f*)vp;           // prefetch (pad tile makes this safe)
        v8f acc = {};
        acc = __builtin_amdgcn_wmma_f32_16x16x4_f32(
            false, Acur, false, Bv, (short)0, acc, false, false);
        // D layout: VGPR j, this lane -> vertex M = j + 8*half, our gaussian.
        const int base = t * 16 + half * 8;
#pragma unroll
        for (int j = 0; j < 8; ++j) {
            float c = acc[j];
            if (c < s3)                 // wave skips bubble when execz
                top4_insert_tail(c, base + j, s0, s1, s2, s3, i0, i1, i2, i3);
        }
    }

    // Merge the two half-lane lists covering the same gaussian.
    {
        float m0 = __shfl_xor(s0, 16, 32);
        float m1 = __shfl_xor(s1, 16, 32);
        float m2 = __shfl_xor(s2, 16, 32);
        float m3 = __shfl_xor(s3, 16, 32);
        int   j0 = __shfl_xor(i0, 16, 32);
        int   j1 = __shfl_xor(i1, 16, 32);
        int   j2 = __shfl_xor(i2, 16, 32);
        int   j3 = __shfl_xor(i3, 16, 32);
        top4_insert(m0, j0, s0, s1, s2, s3, i0, i1, i2, i3);
        top4_insert(m1, j1, s0, s1, s2, s3, i0, i1, i2, i3);
        top4_insert(m2, j2, s0, s1, s2, s3, i0, i1, i2, i3);
        top4_insert(m3, j3, s0, s1, s2, s3, i0, i1, i2, i3);
    }

    float l_mdist = 0.0f, l_base = 0.0f, l_mesh = 0.0f;

    if (lane < 16 && (gbase + lane) < N) {
        const int gg = gbase + lane;
        int idx[4] = { i0, i1, i2, i3 };
        float w[4], wsum = 0.0f;
#pragma unroll
        for (int k = 0; k < 4; ++k) {
            const float* vv = &s_vert[idx[k] << 2];
            float dx = gx - vv[0], dy = gy - vv[1], dz = gz - vv[2];
            float d = sqrtf(dx * dx + dy * dy + dz * dz);
            l_mdist += d;
            float wk = 1.0f / (d + EPSF);
            w[k] = wk;
            wsum += wk;
        }
        const float inv = 1.0f / (wsum + EPSF);

        float a_disp[3] = {0, 0, 0};
        float a_es[2]   = {0, 0};
        float a_q[4]    = {0, 0, 0, 0};
        float a_n[3]    = {0, 0, 0};
        float a_lbs[24];
#pragma unroll
        for (int j = 0; j < 24; ++j) a_lbs[j] = 0.0f;

#pragma unroll
        for (int k = 0; k < 4; ++k) {
            const int ix = idx[k];
            const float wk = w[k] * inv;
            const float* dp = disp + ix * 3;
            a_disp[0] += wk * dp[0];
            a_disp[1] += wk * dp[1];
            a_disp[2] += wk * dp[2];
            const float* sp = vsc + ix * 2;
            a_es[0] += wk * expf(sp[0]);
            a_es[1] += wk * expf(sp[1]);
            const float* qp = vrot + ix * 4;
            float q0 = qp[0], q1 = qp[1], q2 = qp[2], q3 = qp[3];
            float qn = sqrtf(q0 * q0 + q1 * q1 + q2 * q2 + q3 * q3);
            qn = fmaxf(qn, 1e-12f);
            float rq = 1.0f / qn;
            a_q[0] += wk * q0 * rq;
            a_q[1] += wk * q1 * rq;
            a_q[2] += wk * q2 * rq;
            a_q[3] += wk * q3 * rq;
            const float* lp = lbs + ix * 24;
#pragma unroll
            for (int j = 0; j < 24; ++j) a_lbs[j] += wk * lp[j];
            const float* np2 = vnrm + ix * 3;
            a_n[0] += wk * np2[0];
            a_n[1] += wk * np2[1];
            a_n[2] += wk * np2[2];
        }

        // out_xyz
        size_t o = (size_t)gg * 3;
        out[o + 0] = a_disp[0] + gx;
        out[o + 1] = a_disp[1] + gy;
        out[o + 2] = a_disp[2] + gz;
        // out_scaling = log(interp(exp(vs)))
        o = (size_t)N * 3 + (size_t)gg * 2;
        out[o + 0] = logf(a_es[0]);
        out[o + 1] = logf(a_es[1]);
        // out_rotation
        o = (size_t)N * 5 + (size_t)gg * 4;
        out[o + 0] = a_q[0]; out[o + 1] = a_q[1];
        out[o + 2] = a_q[2]; out[o + 3] = a_q[3];
        // out_lbs
        o = (size_t)N * 9 + (size_t)gg * 24;
#pragma unroll
        for (int j = 0; j < 24; ++j) out[o + j] = a_lbs[j];

        // losses
        l_base = fmaxf(a_es[0] - BASE_SCALE_F, 0.0f)
               + fmaxf(a_es[1] - BASE_SCALE_F, 0.0f);
        float qn2 = sqrtf(a_q[0] * a_q[0] + a_q[1] * a_q[1] +
                          a_q[2] * a_q[2] + a_q[3] * a_q[3]);
        float rq2 = 1.0f / qn2;
        float r = a_q[0] * rq2, x = a_q[1] * rq2;
        float y = a_q[2] * rq2, z = a_q[3] * rq2;
        float zx = 2.0f * (x * z + r * y);
        float zy = 2.0f * (y * z - r * x);
        float zz = 1.0f - 2.0f * (x * x + y * y);
        float ex = zx - a_n[0], ey = zy - a_n[1], ez = zz - a_n[2];
        l_mesh = sqrtf(ex * ex + ey * ey + ez * ez);
    }

    // Deterministic block reduction of the 3 loss partials.
#pragma unroll
    for (int off = 16; off > 0; off >>= 1) {
        l_mdist += __shfl_xor(l_mdist, off, 32);
        l_base  += __shfl_xor(l_base,  off, 32);
        l_mesh  += __shfl_xor(l_mesh,  off, 32);
    }
    if (lane == 0) {
        s_red[wid * 3 + 0] = l_mdist;
        s_red[wid * 3 + 1] = l_base;
        s_red[wid * 3 + 2] = l_mesh;
    }
    __syncthreads();
    if (tid == 0) {
        float a = 0, b = 0, c = 0;
        for (int wv = 0; wv < 8; ++wv) {
            a += s_red[wv * 3 + 0];
            b += s_red[wv * 3 + 1];
            c += s_red[wv * 3 + 2];
        }
        part[blockIdx.x * 3 + 0] = a;
        part[blockIdx.x * 3 + 1] = b;
        part[blockIdx.x * 3 + 2] = c;
    }
}

// ---------------------------------------------------------------------------
// Kernel 3: fixed-order finalize of the 4 scalar losses
// ---------------------------------------------------------------------------
__global__ void finalize_kernel(const float* __restrict__ ws,
                                int nb1, int nb2, int N, int V,
                                float* __restrict__ out_losses)
{
    if (blockIdx.x == 0 && threadIdx.x == 0) {
        float dsum = 0.0f;
        for (int i = 0; i < nb1; ++i) dsum += ws[i];
        float md = 0.0f, bs = 0.0f, mn = 0.0f;
        for (int i = 0; i < nb2; ++i) {
            md += ws[32 + i * 3 + 0];
            bs += ws[32 + i * 3 + 1];
            mn += ws[32 + i * 3 + 2];
        }
        out_losses[0] = md / ((float)N * 4.0f);   // loss_mdist
        out_losses[1] = dsum / (float)V;          // loss_disp
        out_losses[2] = bs / ((float)N * 2.0f);   // loss_base_scale
        out_losses[3] = mn / (float)N;            // loss_mesh_normal
    }
}

extern "C" void kernel_launch(void* const* d_in, const int* in_sizes, int n_in,
                              void* d_out, int out_size, void* d_ws, size_t ws_size,
                              hipStream_t stream)
{
    const float* g_xyz = (const float*)d_in[0];
    const float* v_xyz = (const float*)d_in[1];
    const float* disp  = (const float*)d_in[2];
    const float* vsc   = (const float*)d_in[3];
    const float* vrot  = (const float*)d_in[4];
    const float* lbs   = (const float*)d_in[5];
    const float* vnrm  = (const float*)d_in[6];

    const int N = in_sizes[0] / 3;
    const int V = in_sizes[1] / 3;
    const int Vpad = ((V + 15) / 16) * 16;

    float* out = (float*)d_out;
    float* ws  = (float*)d_ws;

    const int nb1 = (V + 255) / 256;          // prep blocks (<=32 assumed)
    const int nb2 = (N + 127) / 128;          // 128 gaussians per block
    const size_t shmem = (size_t)(Vpad + 16) * 4 * sizeof(float);

    prep_disp_kernel<<<nb1, 256, 0, stream>>>(disp, V, ws);
    nn_interp_kernel<<<nb2, 256, shmem, stream>>>(
        g_xyz, v_xyz, disp, vsc, vrot, lbs, vnrm,
        N, V, Vpad, out, ws + 32);
    finalize_kernel<<<1, 64, 0, stream>>>(ws, nb1, nb2, N, V,
                                          out + (size_t)N * 33);
}